// SFI_41008347742360
// MI455X (gfx1250) — compile-verified
//
#include <hip/hip_runtime.h>
#include <hip/hip_bf16.h>

typedef __attribute__((ext_vector_type(2))) float v2f;
typedef __attribute__((ext_vector_type(4))) float v4f;
typedef __attribute__((ext_vector_type(8))) float v8f;

#define Bn   64
#define CDD  5
#define HIS  100
#define TT   30
#define LL   3
#define DD   256
#define KK   10
#define ROW_TOTAL (Bn*CDD + Bn*HIS)          // 6720
#define CDD_ROWS  (Bn*CDD)                   // 320
#define TLD       (TT*LL*DD)                 // 23040 floats per his row
#define OUT_MAIN  ((size_t)Bn*CDD*KK*TLD)    // 73,728,000 floats

// ---------------------------------------------------------------------------
// Kernel A: P = l2norm(X @ W^T + b).  One 16-row strip per block.
// 4 waves/block, each wave computes four 16x16 output tiles via
// v_wmma_f32_16x16x4_f32 over K=256 (64 WMMA steps per tile).
// ---------------------------------------------------------------------------
__global__ __launch_bounds__(128) void proj_norm_kernel(
    const float* __restrict__ cdd, const float* __restrict__ his,
    const float* __restrict__ W,   const float* __restrict__ bsel,
    float* __restrict__ proj)
{
    __shared__ float lds[16 * DD];      // 16 KB strip
    __shared__ float sumsq[16];

    const int tid  = threadIdx.x;
    const int lane = tid & 31;
    const int wave = tid >> 5;
    const int row_base = blockIdx.x * 16;

    // cdd rows [0,320), his rows [320,6720); boundary aligns to tile 20.
    const float* X = (row_base < CDD_ROWS) ? cdd : his;
    const int roff = (row_base < CDD_ROWS) ? row_base : (row_base - CDD_ROWS);

    const int m  = lane & 15;            // row within tile (A) / col within tile (B)
    const int ko = (lane >> 4) * 2;      // K-pair select per ISA A/B f32 layout

    if (tid < 16) sumsq[tid] = 0.0f;

    const float* arow = X + (size_t)(roff + m) * DD + ko;

    #pragma unroll
    for (int ct = 0; ct < 4; ++ct) {
        const int coln = wave * 64 + ct * 16 + m;     // global output column
        const float* brow = W + (size_t)coln * DD + ko;
        v8f acc = {};
        #pragma unroll 8
        for (int k0 = 0; k0 < DD; k0 += 4) {
            v2f a = *(const v2f*)(arow + k0);
            v2f b = *(const v2f*)(brow + k0);
            acc = __builtin_amdgcn_wmma_f32_16x16x4_f32(
                false, a, false, b, (short)0, acc, false, false);
        }
        // C/D layout: lane<16 -> rows 0..7 (col = lane), lane>=16 -> rows 8..15
        const int half = lane >> 4;
        const float bias = bsel[coln];
        #pragma unroll
        for (int i = 0; i < 8; ++i)
            lds[(half * 8 + i) * DD + coln] = acc[i] + bias;
    }
    __syncthreads();

    // Row sum-of-squares: 8 chunks x 16 rows across 128 threads.
    {
        const int row = tid & 15;
        const int c0  = (tid >> 4) * 32;
        float s = 0.0f;
        #pragma unroll
        for (int j = 0; j < 32; ++j) {
            float v = lds[row * DD + c0 + j];
            s += v * v;
        }
        atomicAdd(&sumsq[row], s);
    }
    __syncthreads();

    for (int i = tid; i < 16 * DD; i += 128) {
        const int row = i >> 8;
        const float scale = 1.0f / fmaxf(sqrtf(sumsq[row]), 1e-12f);
        proj[(size_t)(row_base + row) * DD + (i & 255)] = lds[i] * scale;
    }
}

// ---------------------------------------------------------------------------
// Kernel B: attn[b,c,h] = dot(cdd_p[b,c], his_p[b,h]).  One wave per
// (b, 16-wide h tile); M=16 (rows c, 5 valid), N=16 (h), K=256 WMMA.
// ---------------------------------------------------------------------------
__global__ __launch_bounds__(32) void attn_kernel(
    const float* __restrict__ proj, float* __restrict__ attn)
{
    const int b    = blockIdx.x;
    const int nt   = blockIdx.y;            // 0..6 (112 >= 100 cols)
    const int lane = threadIdx.x;
    const int m    = lane & 15;
    const int ko   = (lane >> 4) * 2;

    // A rows: proj rows b*5 + m (rows beyond c=4 are junk, never stored; always
    // in-bounds of the 6720-row proj buffer).
    const float* arow = proj + (size_t)(b * CDD + m) * DD + ko;
    const int h  = nt * 16 + m;
    const int hc = (h < HIS) ? h : (HIS - 1);      // arithmetic clamp, no divergence
    const float* brow = proj + (size_t)(CDD_ROWS + b * HIS + hc) * DD + ko;

    v8f acc = {};
    #pragma unroll 8
    for (int k0 = 0; k0 < DD; k0 += 4) {
        v2f a  = *(const v2f*)(arow + k0);
        v2f bb = *(const v2f*)(brow + k0);
        acc = __builtin_amdgcn_wmma_f32_16x16x4_f32(
            false, a, false, bb, (short)0, acc, false, false);
    }

    const int half = lane >> 4;
    const int hcol = nt * 16 + m;
    if (hcol < HIS) {
        #pragma unroll
        for (int i = 0; i < 8; ++i) {
            const int c = half * 8 + i;
            if (c < CDD)
                attn[(size_t)b * (CDD * HIS) + c * HIS + hcol] = acc[i];
        }
    }
}

// ---------------------------------------------------------------------------
// Kernel C: top-10 of 100 per (b,c).  320 threads total; strict-> argmax
// matches jax.lax.top_k first-occurrence tie-break.
// ---------------------------------------------------------------------------
__global__ __launch_bounds__(64) void topk_kernel(
    const float* __restrict__ attn, float* __restrict__ tkv,
    int* __restrict__ tki, int* __restrict__ idx_out)
{
    const int t = blockIdx.x * blockDim.x + threadIdx.x;
    if (t >= Bn * CDD) return;
    float vals[HIS];
    const float* a = attn + (size_t)t * HIS;
    #pragma unroll 4
    for (int h = 0; h < HIS; ++h) vals[h] = a[h];
    #pragma unroll
    for (int k = 0; k < KK; ++k) {
        float best = -__builtin_huge_valf();
        int bi = 0;
        for (int h = 0; h < HIS; ++h) {
            if (vals[h] > best) { best = vals[h]; bi = h; }
        }
        tkv[t * KK + k] = best;
        tki[t * KK + k] = bi;
        idx_out[t * KK + k] = bi;
        vals[bi] = -__builtin_huge_valf();
    }
}

// ---------------------------------------------------------------------------
// Kernel D: the bandwidth kernel.  One block per selected row (3200 blocks),
// stream 23040 floats as float4 with non-temporal stores (295 MB output >
// 192 MB L2 -> don't cache the writes).
// ---------------------------------------------------------------------------
__global__ __launch_bounds__(256) void gather_scale_kernel(
    const float* __restrict__ emb, const float* __restrict__ tkv,
    const int* __restrict__ tki, float* __restrict__ out)
{
    const int blk = blockIdx.x;                  // b*50 + c*10 + k
    const int b   = blk / (CDD * KK);
    const float v = tkv[blk];
    const float w = (v < 0.1f) ? 0.0f : v;
    const int h   = tki[blk];

    const v4f* __restrict__ src = (const v4f*)(emb + ((size_t)b * HIS + h) * TLD);
    v4f* __restrict__ dst       = (v4f*)(out + (size_t)blk * TLD);

    for (int i = threadIdx.x; i < TLD / 4; i += 256) {
        v4f x = src[i];
        x *= w;
        __builtin_nontemporal_store(x, &dst[i]);
    }
}

// ---------------------------------------------------------------------------
extern "C" void kernel_launch(void* const* d_in, const int* in_sizes, int n_in,
                              void* d_out, int out_size, void* d_ws, size_t ws_size,
                              hipStream_t stream) {
    const float* cdd_repr = (const float*)d_in[0];   // (64,5,256)
    const float* his_repr = (const float*)d_in[1];   // (64,100,256)
    const float* his_emb  = (const float*)d_in[2];   // (64,100,30,3,256)
    const float* W_sel    = (const float*)d_in[3];   // (256,256)
    const float* b_sel    = (const float*)d_in[4];   // (256,)

    // workspace layout
    float* proj = (float*)d_ws;                      // 6720*256 floats
    float* attn = proj + (size_t)ROW_TOTAL * DD;     // 32000 floats
    float* tkv  = attn + (size_t)Bn * CDD * HIS;     // 3200 floats
    int*   tki  = (int*)(tkv + Bn * CDD * KK);       // 3200 ints

    float* out     = (float*)d_out;                  // his_activated (73,728,000 f32)
    int*   idx_out = (int*)(out + OUT_MAIN);         // idx (3200 i32), concatenated

    proj_norm_kernel<<<ROW_TOTAL / 16, 128, 0, stream>>>(
        cdd_repr, his_repr, W_sel, b_sel, proj);

    attn_kernel<<<dim3(Bn, (HIS + 15) / 16), 32, 0, stream>>>(proj, attn);

    topk_kernel<<<(Bn * CDD + 63) / 64, 64, 0, stream>>>(attn, tkv, tki, idx_out);

    gather_scale_kernel<<<Bn * CDD * KK, 256, 0, stream>>>(his_emb, tkv, tki, out);
}